// FNN_54709293416531
// MI455X (gfx1250) — compile-verified
//
#include <hip/hip_runtime.h>
#include <math.h>

typedef __attribute__((ext_vector_type(2))) float v2f;
typedef __attribute__((ext_vector_type(8))) float v8f;

#define SOLVER_EPS 1e-4f
#define SOLVER_ITERS 50

__device__ __forceinline__ float sigmoidf(float x) {
    return 1.0f / (1.0f + expf(-x));
}

// One block = 256 threads = 8 waves (wave32) = 256 rows.
// Stage 1: H = sigmoid(X @ w1) via V_WMMA_F32_16X16X4_F32 (each wave: 2 row-tiles x 2 col-tiles).
// Stage 2: per-thread second layer + 50-iteration fixed-point solver.
__global__ __launch_bounds__(256) void fnn_power_solver_kernel(
    const float* __restrict__ X,   // (B,16)
    const float* __restrict__ w1,  // (16,32)
    const float* __restrict__ w2,  // (32,4)
    float* __restrict__ out,       // (B,4)
    int B)
{
    __shared__ float Hs[256 * 33];   // H tile, padded stride to avoid bank conflicts
    __shared__ float w2s[128];

    const int tid  = threadIdx.x;
    const int lane = tid & 31;
    const int wave = tid >> 5;
    const int blockRow = blockIdx.x * 256;

    if (tid < 128) w2s[tid] = w2[tid];

    // ---------------- Stage 1: WMMA front-end ----------------
    const int halfSel = (lane >> 4) & 1;   // 0: lanes 0-15, 1: lanes 16-31
    const int lmod    = lane & 15;
    const int koff    = halfSel * 2;       // A/B K sub-offset per ISA 16x4 f32 layout

    #pragma unroll
    for (int half = 0; half < 2; ++half) {
        const int lrow0 = wave * 32 + half * 16;   // tile row origin within block
        int row = blockRow + lrow0 + lmod;
        if (row >= B) row = B - 1;                 // safety clamp (B % 256 == 0 in practice)

        // A-matrix (16x4 f32 per WMMA): VGPR0/1 hold K={0,1} (lanes 0-15) or K={2,3} (lanes 16-31)
        v2f a[4];
        #pragma unroll
        for (int k = 0; k < 4; ++k) {
            const float2 av = *reinterpret_cast<const float2*>(X + row * 16 + k * 4 + koff);
            a[k].x = av.x;
            a[k].y = av.y;
        }

        #pragma unroll
        for (int ct = 0; ct < 2; ++ct) {           // two 16-column tiles of the 32 hidden units
            const int col = ct * 16 + lmod;
            v8f c = {};
            #pragma unroll
            for (int k = 0; k < 4; ++k) {          // K = 16 via 4 chained 16x16x4 WMMAs
                v2f b;                              // B-matrix 4x16: mirror of A layout
                b.x = w1[(k * 4 + koff + 0) * 32 + col];
                b.y = w1[(k * 4 + koff + 1) * 32 + col];
                c = __builtin_amdgcn_wmma_f32_16x16x4_f32(
                        false, a[k], false, b, (short)0, c, false, false);
            }
            // C/D layout: VGPR i -> row (i + 8*halfSel), N = lane&15
            #pragma unroll
            for (int i = 0; i < 8; ++i) {
                const int r = lrow0 + i + halfSel * 8;
                Hs[r * 33 + col] = sigmoidf(c[i]);
            }
        }
    }

    __syncthreads();

    // ---------------- Stage 2: per-row tail + solver ----------------
    const int row = blockRow + tid;
    if (row >= B) return;

    float z0 = 0.f, z1 = 0.f, z2 = 0.f, z3 = 0.f;
    const float* h = &Hs[tid * 33];
    #pragma unroll
    for (int j = 0; j < 32; ++j) {
        const float hv = h[j];
        z0 = fmaf(hv, w2s[j * 4 + 0], z0);
        z1 = fmaf(hv, w2s[j * 4 + 1], z1);
        z2 = fmaf(hv, w2s[j * 4 + 2], z2);
        z3 = fmaf(hv, w2s[j * 4 + 3], z3);
    }
    const float y0 = sigmoidf(z0), y1 = sigmoidf(z1);
    const float y2 = sigmoidf(z2), y3 = sigmoidf(z3);

    const float4* Xv = reinterpret_cast<const float4*>(X + row * 16);
    const float4 g1 = Xv[0];   // G1 row-major: [d0, o0, o1, d1]
    const float4 g2 = Xv[1];   // G2 row-major
    const float4 sg = Xv[2];   // sigma1.x, sigma1.y, sigma2.x, sigma2.y
    const float4 rr = Xv[3];   // r_bar0, r_bar1, (r_req unused by solver)

    const float F101 = g1.y / g1.x, F110 = g1.z / g1.w;
    const float F201 = g2.y / g2.x, F210 = g2.z / g2.w;
    const float v10  = sg.x / g1.x, v11  = sg.y / g1.w;
    const float v20  = sg.z / g2.x, v21  = sg.w / g2.w;

    const float is1 = 1.0f / (y0 + y1);
    const float is2 = 1.0f / (y2 + y3);
    const float r10 = rr.x * y0 * is1, r11 = rr.x * y1 * is1;
    const float r20 = rr.y * y2 * is2, r21 = rr.y * y3 * is2;

    float p10 = 0.05f, p11 = 0.2f, p20 = 0.1f, p21 = 0.3f;
    bool done = false;
    for (int it = 0; it < SOLVER_ITERS; ++it) {
        if (done) break;   // reference freezes p one step AFTER err < EPS; p was already updated
        const float s10 = p10 / fmaf(F101, p11, v10);
        const float s11 = p11 / fmaf(F110, p10, v11);
        const float s20 = p20 / fmaf(F201, p21, v20);
        const float s21 = p21 / fmaf(F210, p20, v21);
        const float n10 = r10 * p10 / log1pf(s10);
        const float n11 = r11 * p11 / log1pf(s11);
        const float n20 = r20 * p20 / log1pf(s20);
        const float n21 = r21 * p21 / log1pf(s21);
        const float d10 = n10 - p10, d11 = n11 - p11;
        const float d20 = n20 - p20, d21 = n21 - p21;
        const float err = sqrtf(d10 * d10 + d11 * d11) + sqrtf(d20 * d20 + d21 * d21);
        p10 = n10; p11 = n11; p20 = n20; p21 = n21;
        done = (err < SOLVER_EPS);
    }

    float4 po;
    po.x = p10; po.y = p11; po.z = p20; po.w = p21;
    *reinterpret_cast<float4*>(out + row * 4) = po;
}

extern "C" void kernel_launch(void* const* d_in, const int* in_sizes, int n_in,
                              void* d_out, int out_size, void* d_ws, size_t ws_size,
                              hipStream_t stream) {
    const float* X  = (const float*)d_in[0];
    const float* w1 = (const float*)d_in[1];
    const float* w2 = (const float*)d_in[2];
    float* out = (float*)d_out;

    const int B = in_sizes[0] / 16;
    const int grid = (B + 255) / 256;
    fnn_power_solver_kernel<<<grid, 256, 0, stream>>>(X, w1, w2, out, B);
}